// KMeansClusteringModule_16939351015849
// MI455X (gfx1250) — compile-verified
//
#include <hip/hip_runtime.h>

typedef __attribute__((ext_vector_type(2))) float v2f;
typedef __attribute__((ext_vector_type(8))) float v8f;

#define K_CL 8
#define C_DIM 256
#define N_PTS 4096
#define KM_PASSES 11            // 10 update iterations + final labeling pass
#define WAVES 8
#define THREADS (WAVES * 32)
#define TILE_PTS 16
#define NTILES (N_PTS / TILE_PTS)   // 256 tiles of 16 points
#define LDS_STRIDE 258          // even (b64-aligned) padded stride; A/stage ops <=2-way conflicts,
                                // accumulation reads (c = 2*lane + 64u) fully conflict-free

__global__ __launch_bounds__(THREADS)
void kmeans_cluster_kernel(const float* __restrict__ fm, float* __restrict__ out) {
    // centroids padded to 16 rows; rows 8..15 stay zero (WMMA M-padding)
    __shared__ float cent[16][LDS_STRIDE];
    __shared__ float tile[WAVES][TILE_PTS][LDS_STRIDE];  // per-wave staged point tiles
    __shared__ float sums[K_CL][C_DIM];
    __shared__ float cnorm[K_CL];
    __shared__ unsigned cnt[K_CL];

    const int b    = blockIdx.x;
    const int t    = threadIdx.x;
    const int w    = t >> 5;
    const int lane = t & 31;
    const int m    = lane & 15;   // B column (point-in-tile) / A row (centroid)
    const int ks   = lane >> 4;   // which K-pair this half-wave supplies

    // x[c][i] layout, row stride N_PTS (feature_map[b][c][h][w], i = h*W+w)
    const float* __restrict__ x = fm + (size_t)b * C_DIM * N_PTS;

    // ---- init: cent[k][c] = point k's channel c = x[c*N + k]; zero pad rows ----
    for (int c = t; c < C_DIM; c += THREADS) {
        #pragma unroll
        for (int k = 0; k < K_CL; ++k) cent[k][c] = x[(size_t)c * N_PTS + k];
        #pragma unroll
        for (int k = K_CL; k < 16; ++k) cent[k][c] = 0.0f;
    }
    __syncthreads();

    for (int it = 0; it < KM_PASSES; ++it) {
        // ---- zero accumulators ----
        for (int e = t; e < K_CL * C_DIM; e += THREADS) sums[e >> 8][e & 255] = 0.0f;
        if (t < K_CL) { cnt[t] = 0u; cnorm[t] = 0.0f; }
        __syncthreads();

        // ---- centroid squared norms: wave k reduces cluster k ----
        {
            const int k = t >> 5;
            const int s = t & 31;
            float p = 0.0f;
            #pragma unroll
            for (int u = 0; u < 8; ++u) {
                float cv = cent[k][s + 32 * u];
                p += cv * cv;
            }
            atomicAdd(&cnorm[k], p);
        }
        __syncthreads();

        float cn[K_CL];
        #pragma unroll
        for (int r = 0; r < K_CL; ++r) cn[r] = cnorm[r];

        // per-wave partial sums in VGPRs: acc[k][u] covers channels {2*lane, 2*lane+1} + 64*u
        v2f acc[K_CL][4];
        #pragma unroll
        for (int k = 0; k < K_CL; ++k)
            #pragma unroll
            for (int u = 0; u < 4; ++u) acc[k][u] = (v2f){0.0f, 0.0f};

        // ---- assignment (WMMA f32 16x16x4) + accumulation, per 16-point tile ----
        for (int tl = w; tl < NTILES; tl += WAVES) {
            const int i0 = tl * TILE_PTS;
            v8f d = {};   // D[centroid][point] f32 accumulator

            #pragma unroll 4
            for (int c0 = 0; c0 < C_DIM; c0 += 4) {
                const int ca = c0 + 2 * ks;
                // A (16x4 f32): lane m holds cent[m][ca], cent[m][ca+1]  (ds_load_b64)
                v2f a = *reinterpret_cast<const v2f*>(&cent[m][ca]);
                // B (4x16 f32): lane holds x[ca][i0+m], x[ca+1][i0+m]  (coalesced along points)
                v2f bb;
                bb.x = x[(size_t)ca * N_PTS + i0 + m];
                bb.y = x[(size_t)(ca + 1) * N_PTS + i0 + m];
                // stage tile for the accumulation pass (ds_store_b64, padded stride)
                *reinterpret_cast<v2f*>(&tile[w][m][ca]) = bb;
                d = __builtin_amdgcn_wmma_f32_16x16x4_f32(
                        /*neg_a=*/false, a, /*neg_b=*/false, bb,
                        /*c_mod=*/(short)0, d, /*reuse_a=*/false, /*reuse_b=*/false);
            }

            // lane n<16 holds dot(point i0+n, cent_r) in d[r]; argmin of (||c||^2 - 2*dot)
            float best = 3.4e38f; int bestk = 0;
            #pragma unroll
            for (int r = 0; r < K_CL; ++r) {
                float s = cn[r] - 2.0f * d[r];
                if (s < best) { best = s; bestk = r; }
            }
            if (lane < 16) atomicAdd(&cnt[bestk], 1u);

            // accumulate staged tile into per-wave VGPR partial sums;
            // label is wave-uniform per point -> scalar branch, plain v_add_f32, no atomics
            #pragma unroll 1
            for (int j = 0; j < TILE_PTS; ++j) {
                const int kj = __builtin_amdgcn_readfirstlane(__shfl(bestk, j, 32));
                const float* tp = &tile[w][j][0];
                v2f p0 = *reinterpret_cast<const v2f*>(tp + 2 * lane);
                v2f p1 = *reinterpret_cast<const v2f*>(tp + 2 * lane + 64);
                v2f p2 = *reinterpret_cast<const v2f*>(tp + 2 * lane + 128);
                v2f p3 = *reinterpret_cast<const v2f*>(tp + 2 * lane + 192);
                switch (kj) {
#define ACC_CASE(KK) \
                    case KK: acc[KK][0] += p0; acc[KK][1] += p1; \
                             acc[KK][2] += p2; acc[KK][3] += p3; break;
                    ACC_CASE(0) ACC_CASE(1) ACC_CASE(2) ACC_CASE(3)
                    ACC_CASE(4) ACC_CASE(5) ACC_CASE(6) ACC_CASE(7)
#undef ACC_CASE
                    default: break;
                }
            }
        }

        // flush per-wave partials: 64 ds_add_f32 per wave per pass (conflict-free banks)
        #pragma unroll
        for (int k = 0; k < K_CL; ++k) {
            #pragma unroll
            for (int u = 0; u < 4; ++u) {
                atomicAdd(&sums[k][2 * lane + 64 * u],     acc[k][u].x);
                atomicAdd(&sums[k][2 * lane + 64 * u + 1], acc[k][u].y);
            }
        }
        __syncthreads();

        if (it + 1 < KM_PASSES) {
            // centroid update: cent = cnt>0 ? sums/cnt : cent
            for (int e = t; e < K_CL * C_DIM; e += THREADS) {
                const int k = e >> 8, c = e & 255;
                const unsigned n = cnt[k];
                if (n > 0u) cent[k][c] = sums[k][c] / (float)n;
            }
            __syncthreads();
        } else {
            // final pass: clustered = cnt>0 ? sums/cnt : 0
            float* __restrict__ ob = out + (size_t)b * K_CL * C_DIM;
            for (int e = t; e < K_CL * C_DIM; e += THREADS) {
                const int k = e >> 8, c = e & 255;
                const unsigned n = cnt[k];
                ob[e] = (n > 0u) ? sums[k][c] / (float)n : 0.0f;
            }
        }
    }
}

extern "C" void kernel_launch(void* const* d_in, const int* in_sizes, int n_in,
                              void* d_out, int out_size, void* d_ws, size_t ws_size,
                              hipStream_t stream) {
    (void)n_in; (void)out_size; (void)d_ws; (void)ws_size;
    const float* fm = (const float*)d_in[0];
    float* out = (float*)d_out;
    const int B = in_sizes[0] / (C_DIM * N_PTS);   // 32
    hipLaunchKernelGGL(kmeans_cluster_kernel, dim3(B), dim3(THREADS), 0, stream, fm, out);
}